// KeyedLayer_29265907155540
// MI455X (gfx1250) — compile-verified
//
#include <hip/hip_runtime.h>
#include <stdint.h>

// Problem constants (fixed by the reference).
#define BATCH    64
#define IN_DIM   16384
#define OUT_DIM  16384
#define NNZ      2097152
#define CHUNK    128          // nonzeros per workgroup (NNZ % CHUNK == 0)
#define NBLOCKS  (NNZ / CHUNK)

typedef float v2f __attribute__((ext_vector_type(2)));
typedef float v8f __attribute__((ext_vector_type(8)));

// Metadata word layout (sPack[.].y):
//   bits [19:6]  = col << 6   (pre-shifted element index into xT row base)
//   bits [31:25] = slot       (local distinct-row index, < 128)
#define META_COL_MASK  0x000FFFC0u
#define META_SLOT_SH   25

// -----------------------------------------------------------------------------
// Kernel 1: transpose x [64, 16384] -> xT [16384, 64] so the per-nonzero
// gather of 64 batch values is one contiguous 256B read; also zero the output
// (we accumulate with atomics).
// -----------------------------------------------------------------------------
__global__ __launch_bounds__(256) void prep_kernel(const float* __restrict__ x,
                                                   float* __restrict__ xT,
                                                   float* __restrict__ out) {
    unsigned i = blockIdx.x * 256u + threadIdx.x;   // 0 .. B*IN_DIM-1
    unsigned b = i >> 14;                           // IN_DIM = 2^14
    unsigned c = i & (IN_DIM - 1);
    xT[c * BATCH + b] = x[i];
    out[i] = 0.0f;
}

// -----------------------------------------------------------------------------
// Kernel 2: COO SpMM via V_WMMA_F32_16X16X4_F32.
// Block = 128 threads = 4 waves, owns 128 consecutive sorted-by-row nonzeros.
// Wave w computes batch columns [16w, 16w+16). M-slots = distinct rows in
// chunk (<=16 except with ~1e-13 probability; scalar fallback covers that).
// Packed 16B-aligned LDS records {val, (slot<<25)|(col<<6)} -> one
// ds_load_b128 per K-step; branchless A-build; B address is one v_and_or.
// Chunk staging uses GLOBAL_LOAD_ASYNC_TO_LDS_B32 (ASYNCcnt).
// -----------------------------------------------------------------------------
__global__ __launch_bounds__(128) void spmm_wmma_kernel(const int*   __restrict__ rows,
                                                        const int*   __restrict__ cols,
                                                        const float* __restrict__ vals,
                                                        const float* __restrict__ xT,
                                                        float*       __restrict__ out) {
    __shared__ int   sRow[CHUNK];
    __shared__ int   sCol[CHUNK];
    __shared__ __align__(16) uint2 sPack[CHUNK]; // .x = val bits (async), .y = meta
    __shared__ int   sRowOfSlot[CHUNK];
    __shared__ int   sWaveSum[4];

    const int t    = threadIdx.x;
    const int wave = t >> 5;
    const int lane = t & 31;
    const unsigned gOff = (blockIdx.x * (unsigned)CHUNK + (unsigned)t) * 4u; // byte offset

    // --- async stage: global -> LDS directly (no VGPR round trip) ---
    {
        unsigned ldsRow = (unsigned)(size_t)&sRow[t];
        unsigned ldsCol = (unsigned)(size_t)&sCol[t];
        unsigned ldsVal = (unsigned)(size_t)&sPack[t];   // fills sPack[t].x
        asm volatile("global_load_async_to_lds_b32 %0, %1, %2"
                     :: "v"(ldsRow), "v"(gOff), "s"(rows) : "memory");
        asm volatile("global_load_async_to_lds_b32 %0, %1, %2"
                     :: "v"(ldsCol), "v"(gOff), "s"(cols) : "memory");
        asm volatile("global_load_async_to_lds_b32 %0, %1, %2"
                     :: "v"(ldsVal), "v"(gOff), "s"(vals) : "memory");
        asm volatile("s_wait_asynccnt 0x0" ::: "memory");
    }
    __syncthreads();

    // --- slot index = inclusive scan of row-transition flags (shfl scan) ---
    const int myRow = sRow[t];
    const int flag  = (t > 0 && myRow != sRow[t - 1]) ? 1 : 0;

    int scan = flag;
    #pragma unroll
    for (int d = 1; d < 32; d <<= 1) {
        int up = __shfl_up(scan, d, 32);
        if (lane >= d) scan += up;
    }
    if (lane == 31) sWaveSum[wave] = scan;
    __syncthreads();

    int pre = 0, total = 0;
    #pragma unroll
    for (int w = 0; w < 4; ++w) {
        int s = sWaveSum[w];
        pre   += (w < wave) ? s : 0;
        total += s;
    }
    const int slot      = scan + pre;      // # transitions at or before t
    const int nDistinct = total + 1;

    if (t == 0 || flag) sRowOfSlot[slot] = myRow;
    sPack[t].y = (((unsigned)sCol[t]) << 6) | (((unsigned)slot) << META_SLOT_SH);
    __syncthreads();

    if (nDistinct <= 16) {
        // ---------------- WMMA path (fp32, exact) ----------------
        const int half      = lane >> 4;             // 0: supplies K={0,1}; 1: K={2,3}
        const int l16       = lane & 15;
        const unsigned mTag = ((unsigned)l16) << META_SLOT_SH;  // A-slot tag
        const unsigned nb   = (unsigned)(wave * 16 + l16);      // batch column

        v8f acc = {};
        #pragma unroll
        for (int kk = 0; kk < CHUNK; kk += 4) {
            const int k0 = kk + half * 2;            // even -> 16B aligned
            const uint4 p = *(const uint4*)(&sPack[k0]);  // one ds_load_b128
            // p.x = val0, p.y = meta0, p.z = val1, p.w = meta1
            const float sel0 = ((p.y & 0xFE000000u) == mTag) ? 1.0f : 0.0f;
            const float sel1 = ((p.w & 0xFE000000u) == mTag) ? 1.0f : 0.0f;
            v2f a, b;
            a.x = __uint_as_float(p.x) * sel0;       // unconditional load, branchless
            a.y = __uint_as_float(p.z) * sel1;
            b.x = xT[(p.y & META_COL_MASK) | nb];    // one v_and_or -> saddr+voffset
            b.y = xT[(p.w & META_COL_MASK) | nb];
            acc = __builtin_amdgcn_wmma_f32_16x16x4_f32(
                /*neg_a=*/false, a, /*neg_b=*/false, b,
                /*c_mod=*/(short)0, acc, /*reuse_a=*/false, /*reuse_b=*/false);
        }

        // Flush: D VGPR v holds M = v + half*8, N = l16 (-> batch nb).
        #pragma unroll
        for (int v = 0; v < 8; ++v) {
            const int sm = v + half * 8;
            if (sm < nDistinct) {
                atomicAdd(&out[(size_t)nb * OUT_DIM + sRowOfSlot[sm]], acc[v]);
            }
        }
    } else {
        // ---------------- rare fallback: scalar atomics ----------------
        const unsigned bcol = (unsigned)(t & 63);
        for (int k = (t >> 6); k < CHUNK; k += 2) {
            const uint2 p = sPack[k];
            const float contrib =
                __uint_as_float(p.x) * xT[(p.y & META_COL_MASK) | bcol];
            atomicAdd(&out[(size_t)bcol * OUT_DIM + sRow[k]], contrib);
        }
    }
}

// -----------------------------------------------------------------------------
// Host-side launcher. Inputs (setup_inputs order): x_affine f32[64*16384],
// rows i32[NNZ], cols i32[NNZ], vals f32[NNZ]. Output: f32[64*16384].
// d_ws holds xT (16384*64 floats = 4 MiB).
// -----------------------------------------------------------------------------
extern "C" void kernel_launch(void* const* d_in, const int* in_sizes, int n_in,
                              void* d_out, int out_size, void* d_ws, size_t ws_size,
                              hipStream_t stream) {
    const float* x    = (const float*)d_in[0];
    const int*   rows = (const int*)  d_in[1];
    const int*   cols = (const int*)  d_in[2];
    const float* vals = (const float*)d_in[3];
    float*       out  = (float*)d_out;
    float*       xT   = (float*)d_ws;

    prep_kernel<<<(BATCH * IN_DIM) / 256, 256, 0, stream>>>(x, xT, out);
    spmm_wmma_kernel<<<NBLOCKS, CHUNK, 0, stream>>>(rows, cols, vals, xT, out);
}